// MetaPosterior_23021024707532
// MI455X (gfx1250) — compile-verified
//
#include <hip/hip_runtime.h>

#define D1    1025
#define KK    8
#define PP    4
#define ADIM  1024
#define NROWS (ADIM * PP * KK)   // 32768
#define TPB   256
#define NBLK  (NROWS / TPB)      // 128
#define LOG2PI_F 1.83787706640934548356f

typedef __attribute__((ext_vector_type(2))) float        v2f;
typedef __attribute__((ext_vector_type(8))) float        v8f;
typedef __attribute__((ext_vector_type(4))) unsigned int v4u;
typedef __attribute__((ext_vector_type(8))) int          v8i;
typedef __attribute__((ext_vector_type(4))) int          v4i;

#if defined(__has_builtin)
#  if __has_builtin(__builtin_amdgcn_wmma_f32_16x16x4_f32)
#    define HAVE_WMMA4 1
#  else
#    define HAVE_WMMA4 0
#  endif
#  if __has_builtin(__builtin_amdgcn_tensor_load_to_lds) && __has_builtin(__builtin_amdgcn_s_wait_tensorcnt)
#    define HAVE_TDM 1
#  else
#    define HAVE_TDM 0
#  endif
#else
#  define HAVE_WMMA4 0
#  define HAVE_TDM 0
#endif

// ---------------------------------------------------------------------------
// Wave32 sum reduction via V_WMMA_F32_16X16X4_F32.
// A(16x4): lane L<16 -> A[L,0]=x (VGPR0), A[L,1]=0 (VGPR1);
//          lane L>=16 -> A[L-16,2]=x, A[L-16,3]=0.
// B(4x16) = ones  =>  D[m,n] = x_m + x_{m+16}  (replicated over n).
// Per-lane sum of the 8 C VGPRs gives sum over M=0..7 (lanes 0-15) or
// M=8..15 (lanes 16-31); combine the two halves for the full wave sum.
// EXEC must be all ones (call with the whole block active).
// ---------------------------------------------------------------------------
__device__ __forceinline__ float wave_sum(float x) {
#if HAVE_WMMA4
  v2f a; a.x = x;    a.y = 0.0f;
  v2f b; b.x = 1.0f; b.y = 1.0f;
  v8f c = {};
  c = __builtin_amdgcn_wmma_f32_16x16x4_f32(
      /*neg_a=*/false, a, /*neg_b=*/false, b,
      /*c_mod=*/(short)0, c, /*reuse_a=*/false, /*reuse_b=*/false);
  float t = c[0] + c[1] + c[2] + c[3] + c[4] + c[5] + c[6] + c[7];
  return __shfl(t, 0, 32) + __shfl(t, 16, 32);
#else
  for (int off = 16; off > 0; off >>= 1) x += __shfl_xor(x, off, 32);
  return x;
#endif
}

// ---------------------------------------------------------------------------
// TDM: DMA a 1-D run of f32 from global memory into LDS (ISA ch.7/8 D#).
// Group0: count=1 | lds_addr | global_addr[56:0] | type=2.
// Group1: data_size=4B; tensor_dim0=n; tensor_dim1=1; tile_dim0=n;
//         tensor_dim0_stride=n; no padding / iterate / multicast.
// This toolchain exposes the 6-arg builtin:
//   (v4u g0, v8i g1, v4i g2, v4i g3, v8i extra, i32 cpol)
// ---------------------------------------------------------------------------
#if HAVE_TDM
__device__ __forceinline__ void tdm_load_f32(const float* gptr,
                                             unsigned lds_byte_off,
                                             unsigned nelem) {
  unsigned long long ga = (unsigned long long)(uintptr_t)gptr;
  v4u g0;
  g0[0] = 1u;                                    // count=1 (user descriptor)
  g0[1] = lds_byte_off;                          // lds_addr
  g0[2] = (unsigned)(ga & 0xFFFFFFFFu);          // global_addr[31:0]
  g0[3] = (unsigned)((ga >> 32) & 0x01FFFFFFu)   // global_addr[56:32]
        | (2u << 30);                            // type = 2 ("image")
  v8i g1;
  g1[0] = (int)(2u << 16);                       // data_size=2 -> 4 bytes
  g1[1] = (int)((nelem & 0xFFFFu) << 16);        // tensor_dim0[15:0] @63:48
  g1[2] = (int)(((nelem >> 16) & 0xFFFFu)        // tensor_dim0[31:16]
        | (1u << 16));                           // tensor_dim1 = 1 @95:80
  g1[3] = (int)((nelem & 0xFFFFu) << 16);        // tile_dim0 @127:112
  g1[4] = 0;                                     // tile_dim1=0, tile_dim2=0
  g1[5] = (int)nelem;                            // tensor_dim0_stride[31:0]
  g1[6] = 0;
  g1[7] = 0;
  v4i z4 = {0, 0, 0, 0};                         // groups 2/3 unused (<=2D)
  v8i z8 = {0, 0, 0, 0, 0, 0, 0, 0};
  __builtin_amdgcn_tensor_load_to_lds(g0, g1, z4, z4, z8, /*cpol=*/0);
}
#endif

// ---------------------------------------------------------------------------
// Kernel A: prior term + per-k totals  T_k = sum_j g_k[j]*(theta[j]-m_k[j]).
// One block, 256 threads = 8 waves; wave w owns k = w.
// ws[0..7] = T_k ; ws[8] = (1-1/K)*lp_prior.
// ---------------------------------------------------------------------------
__global__ void __launch_bounds__(TPB)
meta_pre(const float* __restrict__ theta, const float* __restrict__ m_ks,
         const float* __restrict__ grads, const int* __restrict__ alpha_i,
         float* __restrict__ ws) {
  __shared__ float s_red[TPB];
  const int tid  = threadIdx.x;
  const int wave = tid >> 5;
  const int lane = tid & 31;

  float ss = 0.0f;
  for (int j = tid; j < D1; j += TPB) { float t = theta[j]; ss += t * t; }
  s_red[tid] = ss;

  const float* mk = m_ks + wave * D1;
  const float* gk = grads + wave * D1;
  float acc = 0.0f;
  for (int j = lane; j < D1; j += 32) acc += gk[j] * (theta[j] - mk[j]);
  float Tk = wave_sum(acc);                 // whole block active: EXEC all 1s
  if (lane == 0) ws[wave] = Tk;

  __syncthreads();
  if (tid == 0) {
    float s = 0.0f;
    for (int i = 0; i < TPB; ++i) s += s_red[i];
    float alpha = (float)alpha_i[0];
    float lp_prior = -0.5f * ((float)D1 * LOG2PI_F +
                              (float)D1 * __logf(alpha) + s / alpha);
    ws[8] = (1.0f - 1.0f / (float)KK) * lp_prior;
  }
}

// ---------------------------------------------------------------------------
// Kernel B: main pass. Block b owns k = b & 7, chunk = b >> 3.
// Row r = (chunk*TPB + tid)*KK + k ; only perm[0], perm[1] are needed:
//   S = T_k - g0*d0 - g1*d1 ,  d_i = theta[idx_i] - m_k[idx_i]
//   lp = -LOG2PI + 0.5*(log g0 + log g1) - 0.5*(g0*(d0+S)^2 + g1*(d1+S)^2)
// Tables staged into LDS via TDM; block sums via WMMA wave reduction.
// ---------------------------------------------------------------------------
__global__ void __launch_bounds__(TPB)
meta_main(const int* __restrict__ perms, const float* __restrict__ theta,
          const float* __restrict__ m_ks, const float* __restrict__ grads,
          const float* __restrict__ Tk_buf, float* __restrict__ block_sum) {
  __shared__ float smem[3 * D1];            // [theta | m_k | g_k]
  __shared__ float s_wave[TPB / 32];
  float* s_theta = smem;
  float* s_m     = smem + D1;
  float* s_g     = smem + 2 * D1;

  const int tid   = threadIdx.x;
  const int k     = blockIdx.x & (KK - 1);
  const int chunk = blockIdx.x >> 3;

#if HAVE_TDM
  if (tid < 32) {                           // wave 0 issues 3 TDM descriptors
    tdm_load_f32(theta,          0u,                                 D1);
    tdm_load_f32(m_ks + k * D1,  (unsigned)(sizeof(float) * D1),     D1);
    tdm_load_f32(grads + k * D1, (unsigned)(sizeof(float) * 2 * D1), D1);
    __builtin_amdgcn_s_wait_tensorcnt(0);
  }
#else
  for (int j = tid; j < D1; j += TPB) {
    s_theta[j] = theta[j];
    s_m[j]     = m_ks[k * D1 + j];
    s_g[j]     = grads[k * D1 + j];
  }
#endif
  __syncthreads();

  const float Tk = Tk_buf[k];
  const int r = (chunk * TPB + tid) * KK + k;     // row in [0, NROWS)
  const size_t pbase = (size_t)r * D1;
  const int i0 = perms[pbase];
  const int i1 = perms[pbase + 1];

  const float g0 = s_g[i0], g1 = s_g[i1];
  const float d0 = s_theta[i0] - s_m[i0];
  const float d1 = s_theta[i1] - s_m[i1];
  const float S  = Tk - g0 * d0 - g1 * d1;
  const float e0 = d0 + S, e1 = d1 + S;
  const float lp = -LOG2PI_F + 0.5f * (__logf(g0) + __logf(g1))
                 - 0.5f * (g0 * e0 * e0 + g1 * e1 * e1);

  float w = wave_sum(lp);                   // WMMA reduction, EXEC all 1s
  if ((tid & 31) == 0) s_wave[tid >> 5] = w;
  __syncthreads();
  if (tid == 0) {
    float s = 0.0f;
    for (int i = 0; i < TPB / 32; ++i) s += s_wave[i];
    block_sum[blockIdx.x] = s;
  }
}

// ---------------------------------------------------------------------------
// Kernel C: deterministic final combine.
// out = -( ws[8] + (sum of block sums) / (P * M_COND * K) )
// ---------------------------------------------------------------------------
__global__ void meta_final(const float* __restrict__ ws,
                           float* __restrict__ out) {
  float s = 0.0f;
  for (int i = 0; i < NBLK; ++i) s += ws[16 + i];
  out[0] = -(ws[8] + s / (float)(PP * 2 * KK));
}

extern "C" void kernel_launch(void* const* d_in, const int* in_sizes, int n_in,
                              void* d_out, int out_size, void* d_ws,
                              size_t ws_size, hipStream_t stream) {
  const float* theta = (const float*)d_in[0];   // (1025,)  f32
  const float* m_ks  = (const float*)d_in[1];   // (8,1025) f32
  const float* grads = (const float*)d_in[2];   // (8,1025) f32
  const int*   perms = (const int*)d_in[3];     // (1024,4,8,1025) i32
  const int*   alpha = (const int*)d_in[4];     // scalar
  float* ws  = (float*)d_ws;                    // [0..7]=T_k, [8]=loss, [16..]=block sums
  float* out = (float*)d_out;

  meta_pre <<<1,    TPB, 0, stream>>>(theta, m_ks, grads, alpha, ws);
  meta_main<<<NBLK, TPB, 0, stream>>>(perms, theta, m_ks, grads, ws, ws + 16);
  meta_final<<<1,   1,   0, stream>>>(ws, out);
}